// QFFRadianceField_72404558676470
// MI455X (gfx1250) — compile-verified
//
#include <hip/hip_runtime.h>

typedef __attribute__((ext_vector_type(2))) float v2f;
typedef __attribute__((ext_vector_type(8))) float v8f;

#define QN   80
#define NQ2  6400       /* 80*80   */
#define QS   512000     /* 80^3    */

__global__ __launch_bounds__(256, 8)
void qff_radiance_kernel(const float* __restrict__ pos,
                         const float* __restrict__ dir,
                         const float* __restrict__ vol,
                         const float* __restrict__ mlp_w,
                         float* __restrict__ out, int n)
{
    const int tid  = threadIdx.x;
    const int lane = tid & 31;
    const int wid  = tid >> 5;
    int i = blockIdx.x * 256 + tid;
    if (i >= n) i = n - 1;              // keep EXEC all-ones for WMMA

    __shared__ float fbuf[256 * 8];     // 8 KB: per-point MLP outputs

    const float fr[4] = {2.0f, 6.34960420787280f, 20.1587367983188f, 64.0f};
    float p0 = pos[3*i+0], p1 = pos[3*i+1], p2 = pos[3*i+2];

    // ---- B matrix: mlp_w^T (32x8) padded to 4x16 tiles, 32-bit B layout:
    //      vgpr0 = K0 (lanes 0-15) / K2 (lanes 16-31), vgpr1 = K1 / K3
    const int  ncol = lane & 15;
    const bool lo   = lane < 16;
    const int  koff = lo ? 0 : 2;
    float bw[16];
    #pragma unroll
    for (int k = 0; k < 8; ++k) {
        float b0 = 0.0f, b1 = 0.0f;
        if (ncol < 8) {
            b0 = mlp_w[ncol*32 + 4*k + koff];
            b1 = mlp_w[ncol*32 + 4*k + koff + 1];
        }
        bw[2*k] = b0; bw[2*k+1] = b1;
    }

    v8f acc0 = {0,0,0,0,0,0,0,0};       // points wavebase+0..15
    v8f acc1 = {0,0,0,0,0,0,0,0};       // points wavebase+16..31

    // ---- fused: each freq/phase f2 yields one K=4 chunk -> WMMA right away.
    // Keeps live state ~= 2 accumulators + bw[], so we fit 8 waves/SIMD.
    #pragma unroll
    for (int f2 = 0; f2 < 8; ++f2) {
        const float fq = fr[f2 & 3];
        float cx, cy, cz;
        if (f2 < 4) {
            cx = __sinf(p0 * fq); cy = __sinf(p1 * fq); cz = __sinf(p2 * fq);
        } else {
            cx = __cosf(p0 * fq); cy = __cosf(p1 * fq); cz = __cosf(p2 * fq);
        }

        float ux = fminf(fmaxf((cx + 1.0f) * 39.5f, 0.0f), 79.0f);
        float uy = fminf(fmaxf((cy + 1.0f) * 39.5f, 0.0f), 79.0f);
        float uz = fminf(fmaxf((cz + 1.0f) * 39.5f, 0.0f), 79.0f);
        int ix = (int)floorf(ux); if (ix > 78) ix = 78;
        int iy = (int)floorf(uy); if (iy > 78) iy = 78;
        int iz = (int)floorf(uz); if (iz > 78) iz = 78;
        float wx1 = ux - (float)ix, wx0 = 1.0f - wx1;
        float wy1 = uy - (float)iy, wy0 = 1.0f - wy1;
        float wz1 = uz - (float)iz, wz0 = 1.0f - wz1;

        int off = (ix*QN + iy)*QN + iz;
        const float* base = vol + (size_t)f2 * (4 * QS) + off;

        float e[4];
        #pragma unroll
        for (int c = 0; c < 4; ++c) {
            const float* pc = base + c * QS;   // 8 corners, shared offsets
            float v000 = pc[0],        v001 = pc[1];
            float v010 = pc[QN],       v011 = pc[QN + 1];
            float v100 = pc[NQ2],      v101 = pc[NQ2 + 1];
            float v110 = pc[NQ2 + QN], v111 = pc[NQ2 + QN + 1];
            e[c] = ((v000*wz0 + v001*wz1)*wy0 + (v010*wz0 + v011*wz1)*wy1)*wx0
                 + ((v100*wz0 + v101*wz1)*wy0 + (v110*wz0 + v111*wz1)*wy1)*wx1;
        }

        // A(16x4) tiles for both point groups, built with cross-half shuffles
        float q0 = __shfl_xor(e[0], 16, 32);
        float q1 = __shfl_xor(e[1], 16, 32);
        float q2 = __shfl_xor(e[2], 16, 32);
        float q3 = __shfl_xor(e[3], 16, 32);
        v2f a0, a1, b;
        a0.x = lo ? e[0] : q2;  a0.y = lo ? e[1] : q3;
        a1.x = lo ? q0   : e[2]; a1.y = lo ? q1   : e[3];
        b.x  = bw[2*f2];         b.y  = bw[2*f2+1];

        acc0 = __builtin_amdgcn_wmma_f32_16x16x4_f32(false, a0, false, b,
                                                     (short)0, acc0, false, false);
        acc1 = __builtin_amdgcn_wmma_f32_16x16x4_f32(false, a1, false, b,
                                                     (short)0, acc1, false, false);
    }

    // ---- D(16x16) -> LDS so each lane regains its own point's f[0..7] ----
    // D layout: lanes 0-15: vgpr r = row M=r, col=lane; lanes 16-31: M=8+r, col=lane-16
    if (ncol < 8) {
        int pb0 = wid*32 +      (lo ? 0 : 8);
        int pb1 = wid*32 + 16 + (lo ? 0 : 8);
        #pragma unroll
        for (int r = 0; r < 8; ++r) fbuf[(pb0 + r)*8 + ncol] = acc0[r];
        #pragma unroll
        for (int r = 0; r < 8; ++r) fbuf[(pb1 + r)*8 + ncol] = acc1[r];
    }
    __syncthreads();

    float f[8];
    #pragma unroll
    for (int r = 0; r < 8; ++r) f[r] = fbuf[tid*8 + r];

    float dx = dir[3*i+0], dy = dir[3*i+1], dz = dir[3*i+2];
    float df = f[4]*dx + f[5]*dy + f[6]*dz + f[7];
    float r0 = __expf(fminf(f[0] + df, 11.0f));
    float r1 = __expf(fminf(f[1] + df, 11.0f));
    float r2 = __expf(fminf(f[2] + df, 11.0f));
    float av = __expf(fminf(f[3], 11.0f));

    out[3*(size_t)i + 0] = r0;
    out[3*(size_t)i + 1] = r1;
    out[3*(size_t)i + 2] = r2;
    out[3*(size_t)n + i] = av;
}

extern "C" void kernel_launch(void* const* d_in, const int* in_sizes, int n_in,
                              void* d_out, int out_size, void* d_ws, size_t ws_size,
                              hipStream_t stream)
{
    const float* positions  = (const float*)d_in[0];
    const float* directions = (const float*)d_in[1];
    const float* volume     = (const float*)d_in[2];
    const float* mlp_w      = (const float*)d_in[3];
    float* out = (float*)d_out;

    int n = in_sizes[0] / 3;
    int blocks = (n + 255) / 256;
    qff_radiance_kernel<<<blocks, 256, 0, stream>>>(positions, directions, volume,
                                                    mlp_w, out, n);
}